// SphericalHarmonicsAlphaBeta_9010841387427
// MI455X (gfx1250) — compile-verified
//
#include <hip/hip_runtime.h>

// ---------------------------------------------------------------------------
// Real spherical harmonics l=0..8 (81 cols) for 2^20 points.
// Store-bandwidth bound: 324 MB out / 23.3 TB/s ~= 14 us floor.
// Strategy: VALU Horner evaluation -> LDS tile -> async LDS->global DMA
// (CDNA5 global_store_async_from_lds_b128, ASYNCcnt) for linear coalesced
// writes instead of per-thread 324B-strided rows.
// ---------------------------------------------------------------------------

#ifndef __has_builtin
#define __has_builtin(x) 0
#endif

#define AS1 __attribute__((address_space(1)))
#define AS3 __attribute__((address_space(3)))

typedef int v4i __attribute__((ext_vector_type(4)));

namespace shc {

constexpr int LMAX  = 8;
constexpr int NPAIR = 45;   // (l,m), 0<=m<=l<=8
constexpr int MAXT  = 5;    // max Horner terms: (8-0)/2+1

constexpr double cfact(int n) { double r = 1.0; for (int i = 2; i <= n; ++i) r *= i; return r; }
constexpr double ccomb(int n, int k) { return cfact(n) / (cfact(k) * cfact(n - k)); }
constexpr double csqrt(double x) {
  double g = x > 1.0 ? x : 1.0;
  for (int i = 0; i < 64; ++i) g = 0.5 * (g + x / g);
  return g;
}
constexpr double CPI = 3.14159265358979323846;

struct Tables { float c[NPAIR][MAXT]; };

// P_{l,m}(z,y) = y^m * sum_j c[idx][j] * z^{(l-m)-2j}   (Horner in z^2)
// c derived exactly as the reference _legendre_coeffs:
//   norm = sqrt((2l+1)/(4pi) (l-m)!/(l+m)!) / (2^l l!)
//   term k: C(l,k) (-1)^{l-k} (2k)!/(2k-(l+m))! ; power p = 2k-(l+m); k = l-j
constexpr Tables make_tables() {
  Tables t{};
  for (int l = 0; l <= LMAX; ++l) {
    for (int m = 0; m <= l; ++m) {
      const int idx = l * (l + 1) / 2 + m;
      const double norm =
          csqrt((2 * l + 1) / (4.0 * CPI) * cfact(l - m) / cfact(l + m)) /
          ((double)(1 << l) * cfact(l));
      const int nt = (l - m) / 2 + 1;
      for (int j = 0; j < nt; ++j) {
        const int k = l - j;
        const int n = l + m;
        const double c = ccomb(l, k) * ((j & 1) ? -1.0 : 1.0) *
                         (cfact(2 * k) / cfact(2 * k - n)) * norm;
        t.c[idx][j] = (float)c;
      }
    }
  }
  return t;
}

constexpr Tables T = make_tables();

} // namespace shc

constexpr int LMAXK       = 8;
constexpr int NCOL        = 81;                 // (LMAX+1)^2
constexpr int TPB         = 128;                // threads == elements per block
constexpr int TILE_FLOATS = TPB * NCOL;         // 10368 floats = 41472 B LDS
constexpr int TILE_QUADS  = TILE_FLOATS / 4;    // 2592 x 16B

// Async LDS->global 128b store (CDNA5). Tracked by ASYNCcnt.
// Builtin signature (from clang diagnostic): (v4i AS1*, v4i AS3*, imm, imm).
#if __has_builtin(__builtin_amdgcn_global_store_async_from_lds_b128)
#define ASYNC_STORE_B128(gptr, lptr, loff)                                     \
  __builtin_amdgcn_global_store_async_from_lds_b128((AS1 v4i*)(gptr),          \
                                                    (AS3 v4i*)(lptr), 0, 0)
#else
#define ASYNC_STORE_B128(gptr, lptr, loff)                                     \
  asm volatile("global_store_async_from_lds_b128 %0, %1, off" ::"v"(gptr),     \
               "v"(loff)                                                       \
               : "memory")
#endif

__device__ __forceinline__ void wait_asynccnt0() {
#if __has_builtin(__builtin_amdgcn_s_wait_asynccnt)
  __builtin_amdgcn_s_wait_asynccnt(0);
#else
  asm volatile("s_wait_asynccnt 0" ::: "memory");
#endif
}

__global__ __launch_bounds__(TPB) void sh_alpha_beta_kernel(
    const float* __restrict__ alpha, const float* __restrict__ beta,
    float* __restrict__ out, int n) {
  __shared__ float tile[TILE_FLOATS];

  const int tid = threadIdx.x;
  const long long blk_start = (long long)blockIdx.x * TPB;
  long long e = blk_start + tid;
  const long long ec = (e < n) ? e : (long long)n - 1;  // clamp tail reads

  const float a = alpha[ec];
  const float b = beta[ec];

  // z = cos(beta) (Legendre variable), y = sin(beta)
  float y, z, sa, ca;
  __sincosf(b, &y, &z);
  __sincosf(a, &sa, &ca);
  const float z2 = z * z;

  // cos(m a), sin(m a), y^m via angle-addition / product recurrences.
  float cm[LMAXK + 1], sm[LMAXK + 1], yp[LMAXK + 1];
  cm[0] = 1.f; sm[0] = 0.f; yp[0] = 1.f;
  cm[1] = ca;  sm[1] = sa;  yp[1] = y;
#pragma unroll
  for (int m = 2; m <= LMAXK; ++m) {
    cm[m] = cm[m - 1] * ca - sm[m - 1] * sa;
    sm[m] = sm[m - 1] * ca + cm[m - 1] * sa;
    yp[m] = yp[m - 1] * y;
  }

  const float SQ2 = 1.41421356237309504880f;
  float* row = &tile[tid * NCOL];  // stride 81 dwords: 81%64=17 -> conflict-free

#pragma unroll
  for (int l = 0; l <= LMAXK; ++l) {
    const int center = l * l + l;  // column of m=0 (base l^2, offset +l)
#pragma unroll
    for (int m = 0; m <= l; ++m) {
      const int idx = l * (l + 1) / 2 + m;
      const int nt = (l - m) / 2 + 1;
      float acc = shc::T.c[idx][0];
      for (int j = 1; j < nt; ++j) acc = acc * z2 + shc::T.c[idx][j];
      if ((l - m) & 1) acc *= z;
      const float P = acc * yp[m];
      if (m == 0) {
        row[center] = P;
      } else {
        const float t = SQ2 * P;
        row[center + m] = t * cm[m];  // sqrt2*cos(m a)*P
        row[center - m] = t * sm[m];  // sqrt2*sin(m a)*P
      }
    }
  }

  __syncthreads();  // all DS writes visible before the DMA reads them

  // Drain the tile with linear, coalesced async 16B stores.
  float* gbase = out + blk_start * NCOL;
  const long long remf = (long long)n * NCOL - blk_start * NCOL;
  if (remf >= TILE_FLOATS) {
#pragma unroll 1
    for (int q = tid; q < TILE_QUADS; q += TPB) {
      ASYNC_STORE_B128(gbase + 4 * q, tile + 4 * q, (unsigned)(16u * q));
    }
  } else {  // partial tail block (not hit for n = 2^20): scalar fallback
    for (int f = tid; f < (int)remf; f += TPB) gbase[f] = tile[f];
  }
  wait_asynccnt0();
}

extern "C" void kernel_launch(void* const* d_in, const int* in_sizes, int n_in,
                              void* d_out, int out_size, void* d_ws,
                              size_t ws_size, hipStream_t stream) {
  const float* alpha = (const float*)d_in[0];
  const float* beta  = (const float*)d_in[1];
  float* out = (float*)d_out;
  const int n = in_sizes[0];
  const int blocks = (n + TPB - 1) / TPB;
  sh_alpha_beta_kernel<<<blocks, TPB, 0, stream>>>(alpha, beta, out, n);
}